// MI_9096740733041
// MI455X (gfx1250) — compile-verified
//
#include <hip/hip_runtime.h>
#include <math.h>

#define NUM_BINS 64
#define RAD      2
#define NB       68            // NUM_BINS + 2*RAD
#define H_ELEMS  4760          // 68*68 joint + 68 + 68 marginals
#define HP       4800          // padded partial record (floats)
#define SOFF     4624          // source hist offset inside record
#define TOFF     4692          // target hist offset inside record
#define LJ_STRIDE 69           // padded LDS joint row stride (bank decorrelation)
#define TILE_W   128           // voxels per wave per async stage (b128 per lane)
#define WAVES    8             // 256 threads / wave32

typedef float v2f __attribute__((ext_vector_type(2)));
typedef float v8f __attribute__((ext_vector_type(8)));
typedef int   v4i __attribute__((ext_vector_type(4)));

typedef __attribute__((address_space(1))) v4i gv4i;
typedef __attribute__((address_space(3))) v4i lv4i;

// ---------- order-preserving float <-> u32 (for integer atomic min/max) ----------
__device__ __forceinline__ unsigned ordenc(float f) {
  unsigned u = __float_as_uint(f);
  return (u & 0x80000000u) ? ~u : (u | 0x80000000u);
}
__device__ __forceinline__ float orddec(unsigned u) {
  return __uint_as_float((u & 0x80000000u) ? (u & 0x7fffffffu) : ~u);
}

// ---------- cubic B-spline Parzen kernel (sigma = 1) ----------
__device__ __forceinline__ float bsp(float d) {
  float ad = fabsf(d);
  float a2 = ad * ad;
  float r = 0.0f;
  if (ad < 1.0f)       r = (3.0f * a2 * ad - 6.0f * a2 + 4.0f) * (1.0f / 6.0f);
  else if (ad < 2.0f)  { float e = 2.0f - ad; r = e * e * e * (1.0f / 6.0f); }
  return r;
}

// ---------- gfx1250 async global -> LDS staging ----------
__device__ __forceinline__ void async_copy_b128(const float* g, float* l) {
  unsigned loff = (unsigned)(unsigned long long)l;   // low 32 bits of flat LDS ptr == LDS offset
#if __has_builtin(__builtin_amdgcn_global_load_async_to_lds_b128)
  __builtin_amdgcn_global_load_async_to_lds_b128(
      (gv4i*)(unsigned long long)g,
      (lv4i*)(unsigned long long)loff,
      0, 0);
#else
  asm volatile("global_load_async_to_lds_b128 %0, %1, off"
               :: "v"(loff), "v"((unsigned long long)g)
               : "memory");
#endif
}

#if __has_builtin(__builtin_amdgcn_s_wait_asynccnt)
#define WAIT_ASYNC(n) __builtin_amdgcn_s_wait_asynccnt(n)
#else
#define WAIT_ASYNC(n) asm volatile("s_wait_asynccnt " #n ::: "memory")
#endif

// ---------------- K0: init min/max slots ----------------
__global__ void mi_init(unsigned* wsu) {
  int t = threadIdx.x;
  if (t < 4)       wsu[t] = 0xFFFFFFFFu;  // min slots (encoded +inf)
  else if (t < 8)  wsu[t] = 0u;           // max slots (encoded -inf)
}

// ---------------- K1: per-batch min/max of source & target (float4 loads) ----------------
__global__ void mi_minmax(const float* __restrict__ src,
                          const float* __restrict__ tgt,
                          int N, int C, unsigned* wsu) {
  int id  = blockIdx.x;
  int arr = id / (2 * C);
  int rem = id % (2 * C);
  int b   = rem / C;
  int c   = rem % C;
  const float* x = (arr == 0 ? src : tgt) + (long)b * N;
  int N4 = N >> 2;                          // N is a multiple of 4
  int chunk = (N4 + C - 1) / C;
  int beg = c * chunk, fin = min(N4, beg + chunk);
  const float4* x4 = (const float4*)x;
  float mn = 3.4e38f, mx = -3.4e38f;
  for (int i = beg + threadIdx.x; i < fin; i += blockDim.x) {
    float4 v = x4[i];
    mn = fminf(mn, fminf(fminf(v.x, v.y), fminf(v.z, v.w)));
    mx = fmaxf(mx, fmaxf(fmaxf(v.x, v.y), fmaxf(v.z, v.w)));
  }
  __shared__ float smn[256], smx[256];
  smn[threadIdx.x] = mn; smx[threadIdx.x] = mx;
  __syncthreads();
  for (int s = 128; s > 0; s >>= 1) {
    if ((int)threadIdx.x < s) {
      smn[threadIdx.x] = fminf(smn[threadIdx.x], smn[threadIdx.x + s]);
      smx[threadIdx.x] = fmaxf(smx[threadIdx.x], smx[threadIdx.x + s]);
    }
    __syncthreads();
  }
  if (threadIdx.x == 0) {
    atomicMin(&wsu[arr * 2 + b], ordenc(smn[0]));
    atomicMax(&wsu[4 + arr * 2 + b], ordenc(smx[0]));
  }
}

// ---------------- K2: per-block partial histograms (async-staged) ----------------
__global__ void __launch_bounds__(256)
mi_hist(const float* __restrict__ src, const float* __restrict__ tgt,
        int N, int P, int chunk, const unsigned* __restrict__ wsu,
        float* __restrict__ partials) {
  __shared__ float lj[NB * LJ_STRIDE];
  __shared__ float lsh[NB], lth[NB];
  __shared__ float stage[WAVES][2][2][TILE_W];   // [wave][buf][s/t][voxel]

  int tid = threadIdx.x;
  int b = blockIdx.x / P;
  int p = blockIdx.x % P;

  for (int j = tid; j < NB * LJ_STRIDE; j += 256) lj[j] = 0.0f;
  if (tid < NB) { lsh[tid] = 0.0f; lth[tid] = 0.0f; }
  __syncthreads();

  float smn = orddec(wsu[0 + b]), smx = orddec(wsu[4 + b]);
  float tmn = orddec(wsu[2 + b]), tmx = orddec(wsu[6 + b]);
  float sbw = (smx - smn) * (1.0f / NUM_BINS);
  float tbw = (tmx - tmn) * (1.0f / NUM_BINS);
  float spad = smn - sbw * RAD, sinv = 1.0f / sbw;
  float tpad = tmn - tbw * RAD, tinv = 1.0f / tbw;

  const float* sb = src + (long)b * N;
  const float* tb = tgt + (long)b * N;

  int wave = tid >> 5, lane = tid & 31;
  int start = p * chunk;
  int end   = min(N, start + chunk);
  int base0 = start + wave * TILE_W;

  auto stage_tile = [&](int tbase, int buf) {
    int gi = tbase + 4 * lane;
    if (gi + 4 > N) gi = N - 4;            // N is a multiple of 4; keep 16B alignment
    async_copy_b128(sb + gi, &stage[wave][buf][0][4 * lane]);
    async_copy_b128(tb + gi, &stage[wave][buf][1][4 * lane]);
  };

  if (base0 < end) stage_tile(base0, 0);
  int buf = 0;
  for (int t = base0; t < end; t += WAVES * TILE_W, buf ^= 1) {
    int nxt = t + WAVES * TILE_W;
    if (nxt < end) { stage_tile(nxt, buf ^ 1); WAIT_ASYNC(2); }
    else           { WAIT_ASYNC(0); }
#pragma unroll
    for (int q = 0; q < 4; ++q) {
      int gi = t + 4 * lane + q;
      if (gi < end) {
        float sv = stage[wave][buf][0][4 * lane + q];
        float tv = stage[wave][buf][1][4 * lane + q];
        float spos = (sv - spad) * sinv;
        float tpos = (tv - tpad) * tinv;
        float sf = fminf(fmaxf(floorf(spos), 2.0f), 65.0f);
        float tf = fminf(fmaxf(floorf(tpos), 2.0f), 65.0f);
        int smw = (int)sf - 1;             // first window bin, in [1, 64]
        int tmw = (int)tf - 1;
        float sd = spos - (float)smw;
        float td = tpos - (float)tmw;
        float swa[4] = { bsp(sd), bsp(sd - 1.0f), bsp(sd - 2.0f), bsp(sd - 3.0f) };
        float twa[4] = { bsp(td), bsp(td - 1.0f), bsp(td - 2.0f), bsp(td - 3.0f) };
#pragma unroll
        for (int a = 0; a < 4; ++a) {
          atomicAdd(&lsh[smw + a], swa[a]);
          atomicAdd(&lth[tmw + a], twa[a]);
          float* row = &lj[(smw + a) * LJ_STRIDE + tmw];
#pragma unroll
          for (int bb = 0; bb < 4; ++bb) atomicAdd(&row[bb], swa[a] * twa[bb]);
        }
      }
    }
  }
  __syncthreads();

  // flush private partial record (compact 68x68 + marginals)
  float* rec = partials + (long)(b * P + p) * HP;
  for (int j = tid; j < NB * NB; j += 256)
    rec[j] = lj[(j / NB) * LJ_STRIDE + (j % NB)];
  if (tid < NB) rec[SOFF + tid] = lsh[tid];
  if (tid < NB) rec[TOFF + tid] = lth[tid];
}

// ---------------- K3: WMMA tree-reduction across P partials ----------------
// D = A(16x4) * ones(4x16) + C  ==> every D[m][n] = sum_k A[m][k] + C[m][n].
// A[m][k] = partial[p+k][m]: lanes 0-15 load K=0,1 (p,p+1); lanes 16-31 load K=2,3.
__global__ void mi_reduce(const float* __restrict__ partials,
                          float* __restrict__ red, int P) {
  const int G = (H_ELEMS + 15) / 16;       // 298 element groups
  int bb   = blockIdx.x / G;
  int g    = blockIdx.x % G;
  int lane = threadIdx.x;                  // 32 threads = 1 wave, EXEC all ones
  int m    = g * 16 + (lane & 15);
  int mc   = (m < H_ELEMS) ? m : 0;
  float vl = (m < H_ELEMS) ? 1.0f : 0.0f;
  int half = lane >> 4;
  const float* base = partials + (long)bb * P * HP;

  v8f c = {0.f, 0.f, 0.f, 0.f, 0.f, 0.f, 0.f, 0.f};
  v2f bones = {1.0f, 1.0f};
  for (int pp = 0; pp < P; pp += 4) {
    const float* q0 = base + (long)(pp + half * 2) * HP + mc;
    v2f a;
    a.x = q0[0]  * vl;
    a.y = q0[HP] * vl;
    c = __builtin_amdgcn_wmma_f32_16x16x4_f32(false, a, false, bones,
                                              (short)0, c, false, false);
  }
  // D layout: VGPR j -> M=j (lanes 0-15) / M=8+j (lanes 16-31); all N equal.
  if (lane == 0) {
#pragma unroll
    for (int j = 0; j < 8; ++j) {
      int mm = g * 16 + j;
      if (mm < H_ELEMS) red[(long)bb * HP + mm] = c[j];
    }
  } else if (lane == 16) {
#pragma unroll
    for (int j = 0; j < 8; ++j) {
      int mm = g * 16 + 8 + j;
      if (mm < H_ELEMS) red[(long)bb * HP + mm] = c[j];
    }
  }
}

// ---------------- K4: normalize -> densities ----------------
__global__ void mi_finalize(const float* __restrict__ red, float* __restrict__ out) {
  int b = blockIdx.x;
  int tid = threadIdx.x;
  const float* r = red + (long)b * HP;
  __shared__ float sm[256];

  // joint sum
  float acc = 0.0f;
  for (int j = tid; j < NB * NB; j += 256) acc += r[j];
  sm[tid] = acc; __syncthreads();
  for (int s = 128; s > 0; s >>= 1) {
    if (tid < s) sm[tid] += sm[tid + s];
    __syncthreads();
  }
  float jsum = sm[0]; __syncthreads();

  // source sum
  sm[tid] = (tid < NB) ? r[SOFF + tid] : 0.0f; __syncthreads();
  for (int s = 128; s > 0; s >>= 1) {
    if (tid < s) sm[tid] += sm[tid + s];
    __syncthreads();
  }
  float ssum = sm[0]; __syncthreads();

  // target sum
  sm[tid] = (tid < NB) ? r[TOFF + tid] : 0.0f; __syncthreads();
  for (int s = 128; s > 0; s >>= 1) {
    if (tid < s) sm[tid] += sm[tid + s];
    __syncthreads();
  }
  float tsum = sm[0];

  float invs = 1.0f / fmaxf(ssum, 1e-8f);
  float invt = 1.0f / fmaxf(tsum, 1e-8f);
  float invj = 1.0f / fmaxf(jsum, 1e-8f);

  if (tid < NB) out[b * NB + tid]             = r[SOFF + tid] * invs;   // source_density [2,68]
  if (tid < NB) out[2 * NB + b * NB + tid]    = r[TOFF + tid] * invt;   // target_density [2,68]
  for (int j = tid; j < NB * NB; j += 256)
    out[4 * NB + b * NB * NB + j] = r[j] * invj;                        // joint_density [2,68,68]
}

// ---------------- host ----------------
extern "C" void kernel_launch(void* const* d_in, const int* in_sizes, int n_in,
                              void* d_out, int out_size, void* d_ws, size_t ws_size,
                              hipStream_t stream) {
  (void)n_in; (void)out_size;
  const float* src = (const float*)d_in[0];
  const float* tgt = (const float*)d_in[1];
  float* out = (float*)d_out;
  const int B = 2;
  int N = in_sizes[0] / B;                 // 512000 voxels per batch

  unsigned* wsu = (unsigned*)d_ws;
  float*    wsf = (float*)d_ws;

  long cap = (long)(ws_size / 4) - 16 - 2L * HP;
  int P = (int)(cap / (2L * HP));
  if (P > 128) P = 128;
  P &= ~3;
  if (P < 4) P = 4;

  float* partials = wsf + 16;
  float* red      = wsf + 16 + 2L * P * HP;
  int chunk = ((N + P - 1) / P + 1023) & ~1023;   // align per-wave b128 tiles
  const int C = 64;
  const int G = (H_ELEMS + 15) / 16;

  hipLaunchKernelGGL(mi_init,    dim3(1),       dim3(32),  0, stream, wsu);
  hipLaunchKernelGGL(mi_minmax,  dim3(4 * C),   dim3(256), 0, stream, src, tgt, N, C, wsu);
  hipLaunchKernelGGL(mi_hist,    dim3(2 * P),   dim3(256), 0, stream, src, tgt, N, P, chunk, wsu, partials);
  hipLaunchKernelGGL(mi_reduce,  dim3(2 * G),   dim3(32),  0, stream, partials, red, P);
  hipLaunchKernelGGL(mi_finalize,dim3(B),       dim3(256), 0, stream, red, out);
}